// GCN_88613765251764
// MI455X (gfx1250) — compile-verified
//
#include <hip/hip_runtime.h>

typedef __attribute__((ext_vector_type(2))) float v2f;
typedef __attribute__((ext_vector_type(8))) float v8f;

#define GCN_IN    512
#define GCN_HID   256
#define GCN_OUT   128
#define GCN_NODES 50000
#define GCN_EDGES 800000

// Pack B[K][N] row pairs into fragment order: Bp[k/2][col] = {B[k][col], B[k+1][col]}.
// Lets each GEMM lane fetch its whole WMMA B-fragment with one 8-byte load.
__global__ void gcn_pack_b_pairs(const float* __restrict__ B,
                                 float2* __restrict__ Bp,
                                 int K, int N) {
    const int idx = blockIdx.x * blockDim.x + threadIdx.x;   // over (K/2)*N
    if (idx >= (K / 2) * N) return;
    const int k2  = idx / N;
    const int col = idx - k2 * N;
    float2 v;
    v.x = B[(size_t)(2 * k2 + 0) * N + col];
    v.y = B[(size_t)(2 * k2 + 1) * N + col];
    Bp[idx] = v;
}

// C[M,N] = (RELU_A ? relu(A) : A)[M,K] * B[K,N] with V_WMMA_F32_16X16X4_F32.
// One block per 32-row M-tile covering the full N width. The 32xK A slab is
// cooperatively staged into LDS (coalesced float4 loads, padded stride to kill
// bank conflicts; tail rows clamped). Each wave computes a 32x64 strip with 8
// accumulators (2 M-subtiles x 4 N-subtiles): every B fragment feeds 2 WMMAs,
// halving B traffic vs 16-row tiles. B arrives pre-packed in fragment order ->
// one global_load_b64 per fragment. Loop control is wave-uniform; the only
// divergent branches (tail store guards) come after all WMMAs have executed.
template <int K, bool RELU_A>
__global__ void gcn_wmma_gemm_f32(const float* __restrict__ A,
                                  const float2* __restrict__ Bp,   // [K/2][N]
                                  float* __restrict__ C,
                                  int M, int N) {
    constexpr int KP = K + 4;              // padded LDS row stride (floats)
    __shared__ float sA[32 * KP];

    const int tid     = threadIdx.x;
    const int wave    = tid >> 5;
    const int lane    = tid & 31;
    const int rowBase = blockIdx.x * 32;

    // ---- cooperative stage: 32 x K A-tile -> LDS (ReLU fused, rows clamped) ----
    constexpr int NF4R = K / 4;            // float4s per row
    for (int idx = tid; idx < 32 * NF4R; idx += blockDim.x) {
        const int row  = idx / NF4R;
        const int c4   = idx - row * NF4R;
        int grow = rowBase + row;
        if (grow > M - 1) grow = M - 1;    // tail clamp (dup compute, discarded)
        float4 v = ((const float4*)(A + (size_t)grow * K))[c4];
        if (RELU_A) {
            v.x = fmaxf(v.x, 0.f); v.y = fmaxf(v.y, 0.f);
            v.z = fmaxf(v.z, 0.f); v.w = fmaxf(v.w, 0.f);
        }
        *(float4*)&sA[row * KP + 4 * c4] = v;   // 16B-aligned: ds_store_b128
    }
    __syncthreads();

    const int half    = lane >> 4;         // 0: lanes 0-15, 1: lanes 16-31
    const int lr      = lane & 15;
    const int colBase = wave * 64 + lr;    // this wave's 64-col strip

    v8f c00 = {}, c01 = {}, c02 = {}, c03 = {};   // rows rowBase+0..15
    v8f c10 = {}, c11 = {}, c12 = {}, c13 = {};   // rows rowBase+16..31
    for (int k = 0; k < K; k += 4) {
        const int ka = k + 2 * half;       // even
        // A fragments: lane holds rows lr and lr+16, K = ka, ka+1 (LDS b64s)
        const v2f a0 = *(const v2f*)&sA[lr * KP + ka];
        const v2f a1 = *(const v2f*)&sA[(16 + lr) * KP + ka];
        // B fragments: single b64 load each, reused by two WMMAs
        const v2f* brow = (const v2f*)Bp + (size_t)(ka >> 1) * N + colBase;
        v2f b;
        b = brow[0];
        c00 = __builtin_amdgcn_wmma_f32_16x16x4_f32(false, a0, false, b, (short)0, c00, false, false);
        c10 = __builtin_amdgcn_wmma_f32_16x16x4_f32(false, a1, false, b, (short)0, c10, false, false);
        b = brow[16];
        c01 = __builtin_amdgcn_wmma_f32_16x16x4_f32(false, a0, false, b, (short)0, c01, false, false);
        c11 = __builtin_amdgcn_wmma_f32_16x16x4_f32(false, a1, false, b, (short)0, c11, false, false);
        b = brow[32];
        c02 = __builtin_amdgcn_wmma_f32_16x16x4_f32(false, a0, false, b, (short)0, c02, false, false);
        c12 = __builtin_amdgcn_wmma_f32_16x16x4_f32(false, a1, false, b, (short)0, c12, false, false);
        b = brow[48];
        c03 = __builtin_amdgcn_wmma_f32_16x16x4_f32(false, a0, false, b, (short)0, c03, false, false);
        c13 = __builtin_amdgcn_wmma_f32_16x16x4_f32(false, a1, false, b, (short)0, c13, false, false);
    }

    // D layout: VGPR j -> row M = j + 8*half (+subtile), col N = lane&15.
    // Tail guards diverge EXEC only after all WMMAs are done.
    const int r0 = rowBase + half * 8;
#pragma unroll
    for (int j = 0; j < 8; ++j) {
        const int ra = r0 + j;             // M-subtile 0
        if (ra < M) {
            float* p = C + (size_t)ra * N + colBase;
            p[0] = c00[j]; p[16] = c01[j]; p[32] = c02[j]; p[48] = c03[j];
        }
        const int rb = ra + 16;            // M-subtile 1
        if (rb < M) {
            float* p = C + (size_t)rb * N + colBase;
            p[0] = c10[j]; p[16] = c11[j]; p[32] = c12[j]; p[48] = c13[j];
        }
    }
}

// y[dst[e], :] += w[e] * x[src[e], :]   one wave per edge, float4 gather,
// scalar f32 global atomics (targets are L2-resident on the 192MB L2).
template <int F>
__global__ void gcn_spmm_scatter(const int* __restrict__ src,
                                 const int* __restrict__ dst,
                                 const float* __restrict__ w,
                                 const float* __restrict__ x,
                                 float* __restrict__ y,
                                 int nEdges) {
    const int wave = (blockIdx.x * blockDim.x + threadIdx.x) >> 5;
    const int lane = threadIdx.x & 31;
    if (wave >= nEdges) return;            // wave-uniform guard

    const int   s  = src[wave];
    const int   d  = dst[wave];
    const float wt = w[wave];

    const float4* xr = (const float4*)(x + (size_t)s * F);
    float*        yr = y + (size_t)d * F;

#pragma unroll
    for (int i = lane; i < F / 4; i += 32) {
        float4 v = xr[i];
        atomicAdd(&yr[4 * i + 0], wt * v.x);
        atomicAdd(&yr[4 * i + 1], wt * v.y);
        atomicAdd(&yr[4 * i + 2], wt * v.z);
        atomicAdd(&yr[4 * i + 3], wt * v.w);
    }
}

extern "C" void kernel_launch(void* const* d_in, const int* in_sizes, int n_in,
                              void* d_out, int out_size, void* d_ws, size_t ws_size,
                              hipStream_t stream) {
    const float* features = (const float*)d_in[0];   // [50000, 512]
    const int*   esrc     = (const int*)  d_in[1];   // [800000]
    const int*   edst     = (const int*)  d_in[2];   // [800000]
    const float* ew       = (const float*)d_in[3];   // [800000]
    const float* W1       = (const float*)d_in[4];   // [512, 256]
    const float* W2       = (const float*)d_in[5];   // [256, 128]
    float*       out      = (float*)d_out;           // [50000, 128]

    // Workspace carving (xw2 reuses xw1's slot: xw1 is dead after SpMM1).
    float*  xw1 = (float*)d_ws;                        // 50000*256 f32 = 51.2 MB
    float*  h   = xw1 + (size_t)GCN_NODES * GCN_HID;   // 50000*256 f32 = 51.2 MB
    float*  xw2 = xw1;                                 // 50000*128 f32 (aliases xw1)
    float2* W1p = (float2*)(h + (size_t)GCN_NODES * GCN_HID);   // 256*256 f32x2
    float2* W2p = W1p + (size_t)(GCN_IN / 2) * GCN_HID;         // 128*128 f32x2

    const int mBlocks = (GCN_NODES + 31) / 32;         // 1563 (last tile is 16 rows)

    // ---- Pack weights into WMMA fragment order (cheap, deterministic) ----
    gcn_pack_b_pairs<<<(GCN_IN / 2) * GCN_HID / 256, 256, 0, stream>>>(
        W1, W1p, GCN_IN, GCN_HID);
    gcn_pack_b_pairs<<<(GCN_HID / 2) * GCN_OUT / 256, 256, 0, stream>>>(
        W2, W2p, GCN_HID, GCN_OUT);

    // ---- Layer 1: xw1 = features @ W1 ----
    // 1 block per 32-row M-tile; N=256 -> 4 waves of 64 cols -> 128 threads.
    gcn_wmma_gemm_f32<GCN_IN, false><<<mBlocks, 128, 0, stream>>>(
        features, W1p, xw1, GCN_NODES, GCN_HID);

    // ---- SpMM1: h = A @ xw1 (scatter-add); ReLU deferred into GEMM2 stage ----
    hipMemsetAsync(h, 0, (size_t)GCN_NODES * GCN_HID * sizeof(float), stream);
    gcn_spmm_scatter<GCN_HID><<<GCN_EDGES / 8, 256, 0, stream>>>(
        esrc, edst, ew, xw1, h, GCN_EDGES);

    // ---- Layer 2: xw2 = relu(h) @ W2 ----
    // N=128 -> 2 waves of 64 cols -> 64 threads.
    gcn_wmma_gemm_f32<GCN_HID, true><<<mBlocks, 64, 0, stream>>>(
        h, W2p, xw2, GCN_NODES, GCN_OUT);

    // ---- SpMM2: out = A @ xw2 (scatter-add) ----
    hipMemsetAsync(out, 0, (size_t)out_size * sizeof(float), stream);
    gcn_spmm_scatter<GCN_OUT><<<GCN_EDGES / 8, 256, 0, stream>>>(
        esrc, edst, ew, xw2, out, GCN_EDGES);
}